// TPS_76081050681539
// MI455X (gfx1250) — compile-verified
//
#include <hip/hip_runtime.h>

typedef __attribute__((ext_vector_type(2))) float v2f;
typedef __attribute__((ext_vector_type(8))) float v8f;

#define TPS_W    512
#define TPS_H    512
#define TPS_HW   (512 * 512)
#define TPS_B    32
#define TPS_N    64
#define TPS_NP3  67          // N + 3
#define TPS_KPAD 68          // K padded to multiple of 4
#define TPS_NCOL 64          // B * 2 packed output columns
#define WAVES_PER_BLOCK 4

// ---------------------------------------------------------------------------
// Kernel 1: mappingT[k][j] = sum_m inv[k][m] * Y(b, m, d),  j = 2*b + d
//           padded row k == 67 is written as zero.
// ---------------------------------------------------------------------------
__global__ void tps_mapping_kernel(const float* __restrict__ scp,
                                   const float* __restrict__ inv,
                                   const float* __restrict__ pad,
                                   float* __restrict__ mT) {
  for (int t = threadIdx.x; t < TPS_KPAD * TPS_NCOL; t += blockDim.x) {
    const int k = t >> 6;       // 0..67
    const int j = t & 63;       // packed column
    float acc = 0.0f;
    if (k < TPS_NP3) {
      const int b = j >> 1;
      const int d = j & 1;
      for (int m = 0; m < TPS_NP3; ++m) {
        const float y = (m < TPS_N) ? scp[(b * TPS_N + m) * 2 + d]
                                    : pad[(m - TPS_N) * 2 + d];
        acc = fmaf(inv[k * TPS_NP3 + m], y, acc);
      }
    }
    mT[t] = acc;
  }
}

// ---------------------------------------------------------------------------
// Kernel 2: zero-fill mapX/mapY regions of d_out (vectorized b128 stores).
// ---------------------------------------------------------------------------
__global__ void tps_zero_kernel(float4* __restrict__ dst, int n4) {
  const int i = blockIdx.x * blockDim.x + threadIdx.x;
  if (i < n4) dst[i] = make_float4(0.f, 0.f, 0.f, 0.f);
}

// ---------------------------------------------------------------------------
// Kernel 3: sc = repr(HW x 67) @ mappingT(67 x 64) via V_WMMA_F32_16X16X4_F32,
//           plus the map scatter. One 16-row M-tile per wave.
// ---------------------------------------------------------------------------
__global__ void __launch_bounds__(WAVES_PER_BLOCK * 32)
tps_gemm_kernel(const float* __restrict__ repr,
                const float* __restrict__ mT,
                float* __restrict__ out) {
  __shared__ float sB[TPS_KPAD * TPS_NCOL];   // 17 KB, resident per WGP
  for (int t = threadIdx.x; t < TPS_KPAD * TPS_NCOL; t += blockDim.x)
    sB[t] = mT[t];
  __syncthreads();

  const int lane  = threadIdx.x & 31;
  const int wave  = threadIdx.x >> 5;
  const int tile  = blockIdx.x * WAVES_PER_BLOCK + wave;
  const int row0  = tile * 16;                 // first pixel of this M-tile
  const int nloc  = lane & 15;
  const int hi    = lane >> 4;                 // 0 or 1 (lane half)
  const int arow  = row0 + nloc;               // A-matrix row for this lane
  const int khalf = hi * 2;                    // K sub-pair within a chunk

  v8f acc[4] = {};                             // 4 N-tiles of 16x16 f32 C/D

  const float* __restrict__ aptr = repr + (long)arow * TPS_NP3;

  for (int kc = 0; kc < 17; ++kc) {            // K = 68 in chunks of 4
    const int k0 = kc * 4 + khalf;             // this lane's first K
    v2f a;
    a.x = aptr[k0];                                       // k0 <= 66 always
    a.y = (k0 + 1 < TPS_NP3) ? aptr[k0 + 1] : 0.0f;       // guard k == 67
    const float* __restrict__ bbase = &sB[k0 * TPS_NCOL + nloc];
#pragma unroll
    for (int nt = 0; nt < 4; ++nt) {
      v2f b;
      b.x = bbase[nt * 16];                    // B[k0    ][nt*16 + nloc]
      b.y = bbase[nt * 16 + TPS_NCOL];         // B[k0 + 1][nt*16 + nloc]
      acc[nt] = __builtin_amdgcn_wmma_f32_16x16x4_f32(
          /*neg_a=*/false, a, /*neg_b=*/false, b,
          /*c_mod=*/(short)0, acc[nt],
          /*reuse_a=*/false, /*reuse_b=*/false);
    }
  }

  // Epilogue: C/D layout -> lane holds (M = v + 8*hi, N = nloc) per VGPR v.
  const long mapx_base = (long)TPS_B * TPS_HW * 2;
  const long mapy_base = mapx_base + (long)TPS_B * TPS_HW;

#pragma unroll
  for (int nt = 0; nt < 4; ++nt) {
#pragma unroll
    for (int v = 0; v < 8; ++v) {
      const float val = acc[nt][v];
      const int j = nt * 16 + nloc;            // packed column
      const int b = j >> 1;
      const int d = j & 1;
      const int p = row0 + hi * 8 + v;         // pixel index

      // sc output: [B, HW, 2]
      out[((long)b * TPS_HW + p) * 2 + d] = val;

      // Pair lanes (j even <-> j odd, same b,p) to get (x, y) together.
      const float other = __shfl_xor(val, 1, 32);
      if (d == 0) {                            // one scatter per (b, p)
        const float x = val;
        const float y = other;
        const float px = (x + 1.0f) * (0.5f * TPS_W);
        const float py = (y + 1.0f) * (0.5f * TPS_H);
        int idx = (int)(px + (float)TPS_W * py);   // truncate, like astype
        idx = min(max(idx, 0), TPS_HW - 1);
        const float oy = (float)(p >> 9);          // origin_y = p / W
        out[mapx_base + (long)b * TPS_HW + idx] = oy;
        out[mapy_base + (long)b * TPS_HW + idx] = oy;
      }
    }
  }
}

// ---------------------------------------------------------------------------
extern "C" void kernel_launch(void* const* d_in, const int* in_sizes, int n_in,
                              void* d_out, int out_size, void* d_ws, size_t ws_size,
                              hipStream_t stream) {
  const float* scp  = (const float*)d_in[0];   // [B, N, 2]
  const float* inv  = (const float*)d_in[1];   // [67, 67]
  const float* repr = (const float*)d_in[2];   // [HW, 67]
  // d_in[3] (target_coordinate_origin) is derivable: origin_y(p) = p >> 9
  const float* pad  = (const float*)d_in[4];   // [3, 2]
  float* out = (float*)d_out;
  float* mT  = (float*)d_ws;                   // 68*64 floats = 17408 B

  // 1) mappingT (padded) into workspace
  tps_mapping_kernel<<<1, 256, 0, stream>>>(scp, inv, pad, mT);

  // 2) zero mapX/mapY regions (d_out is poisoned by the harness)
  const int n4 = (2 * TPS_B * TPS_HW) / 4;
  tps_zero_kernel<<<(n4 + 255) / 256, 256, 0, stream>>>(
      (float4*)(out + (long)TPS_B * TPS_HW * 2), n4);

  // 3) WMMA GEMM + scatter: 16384 M-tiles, 4 waves/block
  const int ntiles = TPS_HW / 16;
  tps_gemm_kernel<<<ntiles / WAVES_PER_BLOCK, WAVES_PER_BLOCK * 32, 0, stream>>>(
      repr, mT, out);
}